// yoloLoss_80736795230729
// MI455X (gfx1250) — compile-verified
//
#include <hip/hip_runtime.h>
#include <hip/hip_bf16.h>
#include <stdint.h>

// YOLO loss, bandwidth-bound (193 MB @ 23.3 TB/s ~ 8.3 us floor).
// Strategy: CDNA5 async global->LDS B128 copies for coalesced streaming,
// per-cell math from LDS, deterministic two-stage tree reduction.

#define CELLS_PER_BLOCK 256
#define CELL_F 30              // floats per cell
#define CHUNK_F (CELLS_PER_BLOCK * CELL_F)   // 7680 floats = 30720 bytes per buffer
#define CHUNK_V4 (CHUNK_F / 4)               // 1920 float4 per buffer

__device__ __forceinline__ float iou_box(float ax, float ay, float aw, float ah,
                                         float bx, float by, float bw, float bh) {
    float a_x1 = ax - aw * 0.5f, a_x2 = ax + aw * 0.5f;
    float a_y1 = ay - ah * 0.5f, a_y2 = ay + ah * 0.5f;
    float b_x1 = bx - bw * 0.5f, b_x2 = bx + bw * 0.5f;
    float b_y1 = by - bh * 0.5f, b_y2 = by + bh * 0.5f;
    float iw = fmaxf(fminf(a_x2, b_x2) - fmaxf(a_x1, b_x1), 0.0f);
    float ih = fmaxf(fminf(a_y2, b_y2) - fmaxf(a_y1, b_y1), 0.0f);
    float inter = iw * ih;
    float a1 = (a_x2 - a_x1) * (a_y2 - a_y1);
    float a2 = (b_x2 - b_x1) * (b_y2 - b_y1);
    return inter / (a1 + a2 - inter + 1e-6f);
}

__global__ __launch_bounds__(CELLS_PER_BLOCK)
void yolo_loss_cells(const float* __restrict__ pred,
                     const float* __restrict__ target,
                     float* __restrict__ partial) {
    __shared__ alignas(16) float shp[CHUNK_F];
    __shared__ alignas(16) float sht[CHUNK_F];
    __shared__ float red[CELLS_PER_BLOCK];

    const int tid = threadIdx.x;
    const float* gp = pred   + (size_t)blockIdx.x * CHUNK_F;
    const float* gt = target + (size_t)blockIdx.x * CHUNK_F;

    // LDS byte offsets of the staging buffers (low 32 bits of the flat
    // address of an LDS pointer are the LDS offset on CDNA5).
    const unsigned lds_p = (unsigned)(uintptr_t)(void*)shp;
    const unsigned lds_t = (unsigned)(uintptr_t)(void*)sht;

    // Async B128 copy: 1920 float4 per buffer. Seven uniform rounds cover
    // float4 indices [0,1792) for both buffers (each wave instruction moves
    // 32 lanes x 16B = 512B contiguous). Tail round: lanes 0-127 fetch pred's
    // last 128 float4s, lanes 128-255 fetch target's -> all lanes busy.
#pragma unroll
    for (int c = 0; c < 7; ++c) {
        unsigned voff = (unsigned)((c * CELLS_PER_BLOCK + tid) * 16); // bytes
        asm volatile("global_load_async_to_lds_b128 %0, %1, %2"
                     :: "v"(lds_p + voff), "v"(voff), "s"(gp) : "memory");
        asm volatile("global_load_async_to_lds_b128 %0, %1, %2"
                     :: "v"(lds_t + voff), "v"(voff), "s"(gt) : "memory");
    }
    {
        unsigned voff = (unsigned)((7 * CELLS_PER_BLOCK + (tid & 127)) * 16);
        if (tid < 128) {
            asm volatile("global_load_async_to_lds_b128 %0, %1, %2"
                         :: "v"(lds_p + voff), "v"(voff), "s"(gp) : "memory");
        } else {
            asm volatile("global_load_async_to_lds_b128 %0, %1, %2"
                         :: "v"(lds_t + voff), "v"(voff), "s"(gt) : "memory");
        }
    }
    asm volatile("s_wait_asynccnt 0" ::: "memory");
    __syncthreads();

    // ---- per-cell loss (one cell per thread; 30-dword lane stride is
    // bank-conflict-free across 64 banks for a wave32) ----
    const float* p = &shp[tid * CELL_F];
    const float* t = &sht[tid * CELL_F];

    float pv[CELL_F], tv[CELL_F];
#pragma unroll
    for (int k = 0; k < CELL_F; ++k) { pv[k] = p[k]; tv[k] = t[k]; }

    const float p4 = pv[4], p9 = pv[9];
    const float t4 = tv[4], t9 = tv[9];
    const bool obj0 = t4 > 0.0f;   // has_obj (box-0 conf carries the obj flag)
    const bool obj1 = t9 > 0.0f;

    // no-object confidence error (per box, where target conf == 0)
    float d0 = p4 - t4, d1 = p9 - t9;
    float noobj = (obj0 ? 0.0f : d0 * d0) + (obj1 ? 0.0f : d1 * d1);

    // IOUs for both boxes; argmax with first-index tie rule
    float iou0 = iou_box(pv[0], pv[1], pv[2], pv[3], tv[0], tv[1], tv[2], tv[3]);
    float iou1 = iou_box(pv[5], pv[6], pv[7], pv[8], tv[5], tv[6], tv[7], tv[8]);
    const bool sel1 = iou1 > iou0;   // jnp.argmax: index 0 wins ties

    float pbx = sel1 ? pv[5] : pv[0];
    float pby = sel1 ? pv[6] : pv[1];
    float pbw = sel1 ? pv[7] : pv[2];
    float pbh = sel1 ? pv[8] : pv[3];
    float pbc = sel1 ? pv[9] : pv[4];
    float tbx = sel1 ? tv[5] : tv[0];
    float tby = sel1 ? tv[6] : tv[1];
    float tbw = sel1 ? tv[7] : tv[2];
    float tbh = sel1 ? tv[8] : tv[3];
    float tbc = sel1 ? tv[9] : tv[4];

    const float mf = obj0 ? 1.0f : 0.0f;

    float exy = (pbx - tbx) * (pbx - tbx) + (pby - tby) * (pby - tby);

    float pwx = sqrtf(fabsf(pbw) + 1e-6f);
    float pwy = sqrtf(fabsf(pbh) + 1e-6f);
    float twx = sqrtf(obj0 ? tbw : 1.0f);
    float twy = sqrtf(obj0 ? tbh : 1.0f);
    float ewh = (pwx - twx) * (pwx - twx) + (pwy - twy) * (pwy - twy);

    float eobj = (pbc - tbc) * (pbc - tbc);

    float nonbest = (p4 + p9) - pbc;   // conf of the non-best box

    float ecls = 0.0f;
#pragma unroll
    for (int k = 10; k < CELL_F; ++k) {
        float d = pv[k] - tv[k];
        ecls += d * d;
    }

    // total contribution of this cell:
    //   5*coord*mf + obj*mf + 0.5*(noobj_base + 0.5*mf*nonbest^2) + cls*mf
    float cell = mf * (5.0f * (exy + ewh) + eobj + 0.25f * nonbest * nonbest + ecls)
               + 0.5f * noobj;

    // ---- deterministic block tree reduction ----
    red[tid] = cell;
    __syncthreads();
#pragma unroll
    for (int s = CELLS_PER_BLOCK / 2; s > 0; s >>= 1) {
        if (tid < s) red[tid] += red[tid + s];
        __syncthreads();
    }
    if (tid == 0) partial[blockIdx.x] = red[0];
}

__global__ __launch_bounds__(1024)
void yolo_loss_finalize(const float* __restrict__ partial, int nblocks,
                        float scale, float* __restrict__ out) {
    __shared__ float red[1024];
    const int tid = threadIdx.x;
    float s = 0.0f;
    for (int i = tid; i < nblocks; i += 1024) s += partial[i];  // fixed order
    red[tid] = s;
    __syncthreads();
#pragma unroll
    for (int k = 512; k > 0; k >>= 1) {
        if (tid < k) red[tid] += red[tid + k];
        __syncthreads();
    }
    if (tid == 0) out[0] = red[0] * scale;
}

extern "C" void kernel_launch(void* const* d_in, const int* in_sizes, int n_in,
                              void* d_out, int out_size, void* d_ws, size_t ws_size,
                              hipStream_t stream) {
    const float* pred   = (const float*)d_in[0];
    const float* target = (const float*)d_in[1];
    float* out = (float*)d_out;
    float* partial = (float*)d_ws;

    const int n_elems = in_sizes[0];          // 4096*14*14*30
    const int cells   = n_elems / CELL_F;     // 802816
    const int blocks  = cells / CELLS_PER_BLOCK;  // 3136 (exact)
    const int bs      = cells / (14 * 14);    // 4096

    yolo_loss_cells<<<blocks, CELLS_PER_BLOCK, 0, stream>>>(pred, target, partial);
    yolo_loss_finalize<<<1, 1024, 0, stream>>>(partial, blocks,
                                               1.0f / (float)bs, out);
}